// Encoder_12223476925016
// MI455X (gfx1250) — compile-verified
//
#include <hip/hip_runtime.h>
#include <math.h>

#define N_NODES 50000
#define E_EDGES 800000
#define IN_F    128
#define H1_F    128
#define H2_F    64
#define NHEADS  4
#define LATD    32

typedef float v2f __attribute__((ext_vector_type(2)));
typedef float v8f __attribute__((ext_vector_type(8)));

static __device__ __forceinline__ float elu1(float v) {
    return v > 0.f ? v : (__expf(v) - 1.f);
}

// Order-independent float atomic max via monotone integer mapping.
static __device__ __forceinline__ void atomic_max_f32(float* addr, float val) {
    if (val >= 0.f) atomicMax((int*)addr, __float_as_int(val));
    else            atomicMin((unsigned int*)addr, __float_as_uint(val));
}

static __device__ __forceinline__ void atomic_add_f32(float* addr, float val) {
    __hip_atomic_fetch_add(addr, val, __ATOMIC_RELAXED, __HIP_MEMORY_SCOPE_AGENT);
}

// ---------------------------------------------------------------------------
// Generic fp32 WMMA GEMM: C[M,NC] = A[M,K] @ B[K,NC] + bias  (one 16x16 tile/wave)
// A 16x4 layout: lanes 0-15 -> K=k0,k0+1 ; lanes 16-31 -> K=k0+2,k0+3 (M = lane&15)
// B 4x16 layout: mirrored (N = lane&15). C: VGPR j -> row j + 8*(lane>>4).
// ---------------------------------------------------------------------------
__global__ __launch_bounds__(256)
void gemm_bias_wmma(const float* __restrict__ A, const float* __restrict__ B,
                    const float* __restrict__ bias, float* __restrict__ C,
                    int M, int K, int NC, int ldc) {
    int wave = (blockIdx.x * blockDim.x + threadIdx.x) >> 5;
    int lane = threadIdx.x & 31;
    int tilesN = NC >> 4;
    int tileM = wave / tilesN;
    int tileN = wave - tileM * tilesN;
    if (tileM * 16 >= M) return;
    int half = lane >> 4;       // 0 or 1
    int ml   = lane & 15;       // A-row / B-col within tile
    int row0 = tileM << 4, col0 = tileN << 4;

    const float* Arow = A + (size_t)(row0 + ml) * K + 2 * half;
    const float* Bcol = B + (size_t)(2 * half) * NC + (col0 + ml);

    v8f acc = {};
    for (int k0 = 0; k0 < K; k0 += 4) {
        v2f a = *(const v2f*)(Arow + k0);         // A[m][k0+2h], A[m][k0+2h+1]
        v2f b;
        b.x = Bcol[(size_t)k0 * NC];              // B[k0+2h][n]
        b.y = Bcol[(size_t)(k0 + 1) * NC];        // B[k0+2h+1][n]
        acc = __builtin_amdgcn_wmma_f32_16x16x4_f32(false, a, false, b,
                                                    (short)0, acc, false, false);
    }
    float bv = bias ? bias[col0 + ml] : 0.f;
#pragma unroll
    for (int j = 0; j < 8; ++j)
        C[(size_t)(row0 + j + 8 * half) * ldc + col0 + ml] = acc[j] + bv;
}

// ---------------------------------------------------------------------------
// Fused: h0 = ELU(LayerNorm(x @ W1 + b1)).  K = NC = 128.
// One wave owns a 16x128 row stripe (8 accumulator tiles) so each row's full
// 128 columns live in one 16-lane half -> LN reduction via shfl_xor(1..8).
// ---------------------------------------------------------------------------
__global__ __launch_bounds__(256)
void gemm128_ln_elu(const float* __restrict__ A, const float* __restrict__ W,
                    const float* __restrict__ b1, const float* __restrict__ lnw,
                    const float* __restrict__ lnb, float* __restrict__ out, int M) {
    const int K = 128, NC = 128;
    int wave = (blockIdx.x * blockDim.x + threadIdx.x) >> 5;
    int lane = threadIdx.x & 31;
    if (wave * 16 >= M) return;
    int half = lane >> 4, ml = lane & 15;
    int row0 = wave << 4;

    v8f acc[8] = {};
    const float* Arow = A + (size_t)(row0 + ml) * K + 2 * half;
    for (int k0 = 0; k0 < K; k0 += 4) {
        v2f a = *(const v2f*)(Arow + k0);
        const float* Bb = W + (size_t)(k0 + 2 * half) * NC + ml;
#pragma unroll
        for (int t = 0; t < 8; ++t) {
            v2f b;
            b.x = Bb[t * 16];
            b.y = Bb[t * 16 + NC];
            acc[t] = __builtin_amdgcn_wmma_f32_16x16x4_f32(false, a, false, b,
                                                           (short)0, acc[t], false, false);
        }
    }
#pragma unroll
    for (int j = 0; j < 8; ++j) {               // row m = j + 8*half
        float vloc[8], s = 0.f, s2 = 0.f;
#pragma unroll
        for (int t = 0; t < 8; ++t) {
            float v = acc[t][j] + b1[t * 16 + ml];
            vloc[t] = v; s += v; s2 += v * v;
        }
        // reduce over the 16 lanes of this half (xor 1,2,4,8 stays in-half)
#pragma unroll
        for (int off = 1; off < 16; off <<= 1) {
            s  += __shfl_xor(s,  off, 32);
            s2 += __shfl_xor(s2, off, 32);
        }
        float mu   = s  * (1.f / 128.f);
        float var  = s2 * (1.f / 128.f) - mu * mu;
        float rstd = rsqrtf(var + 1e-5f);
        float* orow = out + (size_t)(row0 + j + 8 * half) * NC;
#pragma unroll
        for (int t = 0; t < 8; ++t) {
            int col = t * 16 + ml;
            orow[col] = elu1((vloc[t] - mu) * rstd * lnw[col] + lnb[col]);
        }
    }
}

// ---------------------------------------------------------------------------
// Edge kernels: one thread per (edge, head)
// ---------------------------------------------------------------------------
__global__ __launch_bounds__(256)
void edge_alpha(const float* __restrict__ xl, const float* __restrict__ xr,
                const float* __restrict__ att, const int* __restrict__ ei,
                float* __restrict__ alpha, float* __restrict__ mbuf) {
    int idx = blockIdx.x * blockDim.x + threadIdx.x;
    if (idx >= E_EDGES * NHEADS) return;
    int e = idx >> 2, h = idx & 3;
    int s = ei[e], d = ei[E_EDGES + e];
    const float4* pl = (const float4*)(xl + (size_t)s * (NHEADS * H2_F) + h * H2_F);
    const float4* pr = (const float4*)(xr + (size_t)d * (NHEADS * H2_F) + h * H2_F);
    const float4* pa = (const float4*)(att + h * H2_F);
    float acc = 0.f;
#pragma unroll
    for (int c = 0; c < H2_F / 4; ++c) {
        float4 a4 = pl[c], b4 = pr[c], w4 = pa[c];
        float v;
        v = a4.x + b4.x; v = v > 0.f ? v : 0.2f * v; acc += v * w4.x;
        v = a4.y + b4.y; v = v > 0.f ? v : 0.2f * v; acc += v * w4.y;
        v = a4.z + b4.z; v = v > 0.f ? v : 0.2f * v; acc += v * w4.z;
        v = a4.w + b4.w; v = v > 0.f ? v : 0.2f * v; acc += v * w4.w;
    }
    alpha[idx] = acc;
    atomic_max_f32(&mbuf[(size_t)d * NHEADS + h], acc);
}

__global__ __launch_bounds__(256)
void edge_softmax(float* __restrict__ alpha, const float* __restrict__ mbuf,
                  float* __restrict__ dbuf, const int* __restrict__ ei) {
    int idx = blockIdx.x * blockDim.x + threadIdx.x;
    if (idx >= E_EDGES * NHEADS) return;
    int e = idx >> 2, h = idx & 3;
    int d = ei[E_EDGES + e];
    float ea = __expf(alpha[idx] - mbuf[(size_t)d * NHEADS + h]);
    alpha[idx] = ea;
    atomic_add_f32(&dbuf[(size_t)d * NHEADS + h], ea);
}

// head-mean folded in: hout[d, c] += 0.25 * a * xl[s, h, c]
__global__ __launch_bounds__(256)
void edge_aggregate(const float* __restrict__ alpha, const float* __restrict__ dbuf,
                    const float* __restrict__ xl, const int* __restrict__ ei,
                    float* __restrict__ hout) {
    int idx = blockIdx.x * blockDim.x + threadIdx.x;
    if (idx >= E_EDGES * NHEADS) return;
    int e = idx >> 2, h = idx & 3;
    int s = ei[e], d = ei[E_EDGES + e];
    float a = alpha[idx] / (dbuf[(size_t)d * NHEADS + h] + 1e-16f) * 0.25f;
    const float4* pl = (const float4*)(xl + (size_t)s * (NHEADS * H2_F) + h * H2_F);
    float* orow = hout + (size_t)d * H2_F;
#pragma unroll
    for (int c = 0; c < H2_F / 4; ++c) {
        float4 v = pl[c];
        atomic_add_f32(&orow[4 * c + 0], a * v.x);
        atomic_add_f32(&orow[4 * c + 1], a * v.y);
        atomic_add_f32(&orow[4 * c + 2], a * v.z);
        atomic_add_f32(&orow[4 * c + 3], a * v.w);
    }
}

__global__ __launch_bounds__(256)
void bias_elu(float* __restrict__ h, const float* __restrict__ bias,
              const float* __restrict__ resid, size_t n) {
    size_t i = (size_t)blockIdx.x * blockDim.x + threadIdx.x;
    if (i >= n) return;
    float v = h[i] + bias[i & (H2_F - 1)] + (resid ? resid[i] : 0.f);
    h[i] = elu1(v);
}

__global__ __launch_bounds__(256)
void fill_f32(float* __restrict__ p, float v, size_t n) {
    size_t i = (size_t)blockIdx.x * blockDim.x + threadIdx.x;
    if (i < n) p[i] = v;
}

// ---------------------------------------------------------------------------
extern "C" void kernel_launch(void* const* d_in, const int* in_sizes, int n_in,
                              void* d_out, int out_size, void* d_ws, size_t ws_size,
                              hipStream_t stream) {
    (void)in_sizes; (void)n_in; (void)out_size; (void)ws_size;
    const float* x     = (const float*)d_in[0];
    const int*   ei    = (const int*)  d_in[1];
    const float* W1    = (const float*)d_in[2];
    const float* b1    = (const float*)d_in[3];
    const float* lnw   = (const float*)d_in[4];
    const float* lnb   = (const float*)d_in[5];
    const float* Wl1   = (const float*)d_in[6];
    const float* bl1   = (const float*)d_in[7];
    const float* Wr1   = (const float*)d_in[8];
    const float* br1   = (const float*)d_in[9];
    const float* att1  = (const float*)d_in[10];
    const float* bias1 = (const float*)d_in[11];
    const float* Wl2   = (const float*)d_in[12];
    const float* bl2   = (const float*)d_in[13];
    const float* Wr2   = (const float*)d_in[14];
    const float* br2   = (const float*)d_in[15];
    const float* att2  = (const float*)d_in[16];
    const float* bias2 = (const float*)d_in[17];
    const float* Wmu   = (const float*)d_in[18];
    const float* bmu   = (const float*)d_in[19];
    const float* Wlv   = (const float*)d_in[20];
    const float* blv   = (const float*)d_in[21];

    // Workspace carve (floats). h1/h2 alias h0's region (h0 dead after xl/xr GEMMs).
    float* ws    = (float*)d_ws;
    float* h0    = ws;                                  // N*128
    float* h1    = ws;                                  // N*64  (aliases h0 lo)
    float* h2    = ws + (size_t)N_NODES * H2_F;         // N*64  (aliases h0 hi)
    float* xl    = ws + (size_t)N_NODES * H1_F;         // N*256
    float* xr    = xl + (size_t)N_NODES * NHEADS * H2_F;// N*256
    float* alpha = xr + (size_t)N_NODES * NHEADS * H2_F;// E*4
    float* mbuf  = alpha + (size_t)E_EDGES * NHEADS;    // N*4
    float* dbuf  = mbuf + (size_t)N_NODES * NHEADS;     // N*4

    // 1. h0 = ELU(LN(x @ W1 + b1))   (3125 waves)
    {
        int waves  = N_NODES / 16;
        int blocks = (waves * 32 + 255) / 256;
        gemm128_ln_elu<<<blocks, 256, 0, stream>>>(x, W1, b1, lnw, lnb, h0, N_NODES);
    }
    auto gemm = [&](const float* A, const float* B, const float* bias, float* C,
                    int K, int NC, int ldc) {
        int tiles  = (N_NODES / 16) * (NC / 16);
        int blocks = (tiles + 7) / 8;   // 8 waves (tiles) per 256-thread block
        gemm_bias_wmma<<<blocks, 256, 0, stream>>>(A, B, bias, C, N_NODES, K, NC, ldc);
    };
    auto run_edges = [&](const float* att, float* hout) {
        size_t n4 = (size_t)N_NODES * NHEADS;
        size_t nh = (size_t)N_NODES * H2_F;
        fill_f32<<<(int)((n4 + 255) / 256), 256, 0, stream>>>(mbuf, -__builtin_huge_valf(), n4);
        fill_f32<<<(int)((n4 + 255) / 256), 256, 0, stream>>>(dbuf, 0.f, n4);
        fill_f32<<<(int)((nh + 255) / 256), 256, 0, stream>>>(hout, 0.f, nh);
        int et = E_EDGES * NHEADS, eb = (et + 255) / 256;
        edge_alpha    <<<eb, 256, 0, stream>>>(xl, xr, att, ei, alpha, mbuf);
        edge_softmax  <<<eb, 256, 0, stream>>>(alpha, mbuf, dbuf, ei);
        edge_aggregate<<<eb, 256, 0, stream>>>(alpha, dbuf, xl, ei, hout);
    };

    // GAT layer 1
    gemm(h0, Wl1, bl1, xl, H1_F, NHEADS * H2_F, NHEADS * H2_F);
    gemm(h0, Wr1, br1, xr, H1_F, NHEADS * H2_F, NHEADS * H2_F);
    run_edges(att1, h1);
    {
        size_t nh = (size_t)N_NODES * H2_F;
        bias_elu<<<(int)((nh + 255) / 256), 256, 0, stream>>>(h1, bias1, nullptr, nh);
    }
    // GAT layer 2 (+ residual)
    gemm(h1, Wl2, bl2, xl, H2_F, NHEADS * H2_F, NHEADS * H2_F);
    gemm(h1, Wr2, br2, xr, H2_F, NHEADS * H2_F, NHEADS * H2_F);
    run_edges(att2, h2);
    {
        size_t nh = (size_t)N_NODES * H2_F;
        bias_elu<<<(int)((nh + 255) / 256), 256, 0, stream>>>(h2, bias2, h1, nh);
    }
    // Heads: mu, log_var -> d_out (concatenated)
    float* outf = (float*)d_out;
    gemm(h2, Wmu, bmu, outf, H2_F, LATD, LATD);
    gemm(h2, Wlv, blv, outf + (size_t)N_NODES * LATD, H2_F, LATD, LATD);
}